// Model_49890340110355
// MI455X (gfx1250) — compile-verified
//
#include <hip/hip_runtime.h>
#include <hip/hip_bf16.h>

#define N_NODES 100000
#define NREL    4
#define NEDGE   150000
#define DIN     128
#define DH      256
#define DOUT    128

typedef float v2f __attribute__((ext_vector_type(2)));
typedef float v8f __attribute__((ext_vector_type(8)));

// Native CDNA5 fp32 global atomic add (no-return form -> STOREcnt, L2 atomic unit)
__device__ __forceinline__ void gatomic_add_f32(float* p, float v) {
    asm volatile("global_atomic_add_f32 %0, %1, off" :: "v"(p), "v"(v) : "memory");
}

// ---------------- degree counting ----------------
__global__ __launch_bounds__(256)
void deg_count_kernel(const int* __restrict__ idx, float* __restrict__ deg,
                      int total, int epr, int n) {
    int i = blockIdx.x * 256 + threadIdx.x;
    if (i >= total) return;
    int r = i / epr;
    gatomic_add_f32(&deg[(size_t)r * n + idx[i]], 1.0f);
}

__global__ __launch_bounds__(256)
void rsqrt_kernel(float* __restrict__ c, int total) {
    int i = blockIdx.x * 256 + threadIdx.x;
    if (i >= total) return;
    float d = c[i];
    c[i] = (d > 0.0f) ? (1.0f / sqrtf(d)) : 0.0f;
}

// h[n,f] = sum_r b[r,f]  (bias broadcast over all nodes)
__global__ __launch_bounds__(256)
void bias_init_kernel(float* __restrict__ h, const float* __restrict__ b,
                      long total, int f) {
    long i = (long)blockIdx.x * 256 + threadIdx.x;
    if (i >= total) return;
    int col = (int)(i % f);
    float s = 0.0f;
#pragma unroll
    for (int r = 0; r < NREL; ++r) s += b[r * f + col];
    h[i] = s;
}

// ---------------- fp32 WMMA GEMM: C[nrows,F] = op(A[nrows,K]) @ W[K,F] ----------------
// Block tile 64x128, 8 waves as 2x4; each wave owns a 32x32 tile = 2x2 WMMA
// accumulators (32 WMMAs per 32-K chunk per wave -> high wmma:ds density).
template<int K, int F, bool RELU>
__global__ __launch_bounds__(256)
void gemm_kernel(const float* __restrict__ A, const float* __restrict__ W,
                 float* __restrict__ C, int nrows) {
    constexpr int LDA = 36;   // padded (floats): conflict-free frag reads, 16B-aligned rows
    constexpr int LDB = 132;  // 128 + 4 pad: 16B-aligned rows
    __shared__ float As[64 * LDA];
    __shared__ float Bs[32 * LDB];

    const int t    = threadIdx.x;
    const int lane = t & 31;
    const int wave = t >> 5;
    const int lid  = lane & 15;
    const int koff = (lane < 16) ? 0 : 2;   // fp32 frags: lanes 16-31 hold K+2,K+3
    const int wm   = wave >> 2;             // 0..1  (wave-row)
    const int wn   = wave & 3;              // 0..3  (wave-col)
    const int row0 = blockIdx.x * 64;
    const int col0 = blockIdx.y * 128;

    v8f acc[2][2];
#pragma unroll
    for (int i = 0; i < 2; ++i)
#pragma unroll
        for (int j = 0; j < 2; ++j)
            acc[i][j] = (v8f){0.f, 0.f, 0.f, 0.f, 0.f, 0.f, 0.f, 0.f};

    for (int k0 = 0; k0 < K; k0 += 32) {
        // stage A tile (64x32): 512 float4 slots, 2 per thread, coalesced
#pragma unroll
        for (int i = 0; i < 2; ++i) {
            int s  = t + i * 256;
            int r  = s >> 3;
            int c4 = (s & 7) << 2;
            int gr = row0 + r;
            float4 v = {0.f, 0.f, 0.f, 0.f};
            if (gr < nrows)
                v = *(const float4*)(A + (size_t)gr * K + k0 + c4);
            if (RELU) {
                v.x = fmaxf(v.x, 0.f); v.y = fmaxf(v.y, 0.f);
                v.z = fmaxf(v.z, 0.f); v.w = fmaxf(v.w, 0.f);
            }
            *(float4*)(&As[r * LDA + c4]) = v;
        }
        // stage B tile (32x128): 1024 float4 slots, 4 per thread, coalesced
#pragma unroll
        for (int i = 0; i < 4; ++i) {
            int s  = t + i * 256;
            int r  = s >> 5;
            int c4 = (s & 31) << 2;
            float4 v = *(const float4*)(W + (size_t)(k0 + r) * F + col0 + c4);
            *(float4*)(&Bs[r * LDB + c4]) = v;
        }
        __syncthreads();

#pragma unroll
        for (int kk = 0; kk < 32; kk += 4) {
            v2f a[2], b[2];
#pragma unroll
            for (int i = 0; i < 2; ++i) {
                const float* ap = &As[(wm * 32 + i * 16 + lid) * LDA + kk + koff];
                a[i][0] = ap[0];
                a[i][1] = ap[1];
            }
#pragma unroll
            for (int j = 0; j < 2; ++j) {
                const float* bp = &Bs[(kk + koff) * LDB + wn * 32 + j * 16 + lid];
                b[j][0] = bp[0];
                b[j][1] = bp[LDB];
            }
#pragma unroll
            for (int i = 0; i < 2; ++i)
#pragma unroll
                for (int j = 0; j < 2; ++j)
                    acc[i][j] = __builtin_amdgcn_wmma_f32_16x16x4_f32(
                        false, a[i], false, b[j], (short)0, acc[i][j], false, false);
        }
        __syncthreads();
    }

    // store: VGPR q -> M = q (lanes 0-15) / q+8 (lanes 16-31), N = lid
    const int mbase = row0 + wm * 32;
    const int nbase = col0 + wn * 32;
#pragma unroll
    for (int i = 0; i < 2; ++i) {
        int crow0 = mbase + i * 16 + ((lane < 16) ? 0 : 8);
#pragma unroll
        for (int j = 0; j < 2; ++j) {
            int ccol = nbase + j * 16 + lid;
#pragma unroll
            for (int q = 0; q < 8; ++q) {
                int r = crow0 + q;
                if (r < nrows) C[(size_t)r * F + ccol] = acc[i][j][q];
            }
        }
    }
}

// ---------------- per-edge scatter: h[dst] += hW[src] * c_src[src]*c_dst[dst] ----------------
template<int F>
__global__ __launch_bounds__(256)
void scatter_kernel(const float* __restrict__ hW, const int* __restrict__ src,
                    const int* __restrict__ dst, const float* __restrict__ csrc,
                    const float* __restrict__ cdst, float* __restrict__ h, int E) {
    int w    = blockIdx.x * 8 + (threadIdx.x >> 5);   // one wave per edge
    int lane = threadIdx.x & 31;
    if (w >= E) return;
    int s = src[w], d = dst[w];
    float scale = csrc[s] * cdst[d];
    const float4* sp = (const float4*)(hW + (size_t)s * F);
    float* dp = h + (size_t)d * F;
#pragma unroll
    for (int i = 0; i < F / 128; ++i) {
        float4 v = sp[lane + i * 32];
        int f = (lane + i * 32) * 4;
        gatomic_add_f32(dp + f + 0, v.x * scale);
        gatomic_add_f32(dp + f + 1, v.y * scale);
        gatomic_add_f32(dp + f + 2, v.z * scale);
        gatomic_add_f32(dp + f + 3, v.w * scale);
    }
}

// ---------------- edge dot product over 128 features ----------------
__global__ __launch_bounds__(256)
void dot_kernel(const float* __restrict__ h2, const int* __restrict__ src,
                const int* __restrict__ dst, const int* __restrict__ etype,
                float* __restrict__ out, int E) {
    int w    = blockIdx.x * 8 + (threadIdx.x >> 5);
    int lane = threadIdx.x & 31;
    if (w >= E) return;
    int et = etype[0];
    int s = src[(size_t)et * E + w];
    int d = dst[(size_t)et * E + w];
    float4 a = ((const float4*)(h2 + (size_t)s * DOUT))[lane];
    float4 b = ((const float4*)(h2 + (size_t)d * DOUT))[lane];
    float p = a.x * b.x + a.y * b.y + a.z * b.z + a.w * b.w;
#pragma unroll
    for (int off = 16; off > 0; off >>= 1)
        p += __shfl_down(p, off, 32);
    if (lane == 0) out[w] = p;
}

extern "C" void kernel_launch(void* const* d_in, const int* in_sizes, int n_in,
                              void* d_out, int out_size, void* d_ws, size_t ws_size,
                              hipStream_t stream) {
    const float* x     = (const float*)d_in[0];
    const float* W1    = (const float*)d_in[1];
    const float* b1    = (const float*)d_in[2];
    const float* W2    = (const float*)d_in[3];
    const float* b2    = (const float*)d_in[4];
    const int*   src   = (const int*)d_in[5];
    const int*   dst   = (const int*)d_in[6];
    const int*   etype = (const int*)d_in[7];
    float*       out   = (float*)d_out;

    float* ws = (float*)d_ws;
    size_t off = 0;
    float* c_src = ws + off; off += (size_t)NREL * N_NODES;
    float* c_dst = ws + off; off += (size_t)NREL * N_NODES;
    float* h1    = ws + off; off += (size_t)N_NODES * DH;
    float* h2    = ws + off; off += (size_t)N_NODES * DOUT;
    float* hW    = ws + off; off += (size_t)N_NODES * DH;
    (void)ws_size; (void)in_sizes; (void)n_in; (void)out_size;

    // degrees -> 1/sqrt
    hipMemsetAsync(c_src, 0, 2ull * NREL * N_NODES * sizeof(float), stream);
    {
        int total = NREL * NEDGE;
        int blocks = (total + 255) / 256;
        deg_count_kernel<<<blocks, 256, 0, stream>>>(src, c_src, total, NEDGE, N_NODES);
        deg_count_kernel<<<blocks, 256, 0, stream>>>(dst, c_dst, total, NEDGE, N_NODES);
        int ct = 2 * NREL * N_NODES;
        rsqrt_kernel<<<(ct + 255) / 256, 256, 0, stream>>>(c_src, ct);
    }

    // ---- layer 1: h1 = sum_r scatter(x @ W1[r]) + sum_r b1[r] ----
    {
        long total = (long)N_NODES * DH;
        bias_init_kernel<<<(unsigned)((total + 255) / 256), 256, 0, stream>>>(h1, b1, total, DH);
    }
    dim3 g1((N_NODES + 63) / 64, DH / 128);
    int eblocks = (NEDGE + 7) / 8;
    for (int r = 0; r < NREL; ++r) {
        gemm_kernel<DIN, DH, false><<<g1, 256, 0, stream>>>(
            x, W1 + (size_t)r * DIN * DH, hW, N_NODES);
        scatter_kernel<DH><<<eblocks, 256, 0, stream>>>(
            hW, src + (size_t)r * NEDGE, dst + (size_t)r * NEDGE,
            c_src + (size_t)r * N_NODES, c_dst + (size_t)r * N_NODES, h1, NEDGE);
    }

    // ---- layer 2 (ReLU fused into GEMM A-load): h2 = sum_r scatter(relu(h1) @ W2[r]) + b2 ----
    {
        long total = (long)N_NODES * DOUT;
        bias_init_kernel<<<(unsigned)((total + 255) / 256), 256, 0, stream>>>(h2, b2, total, DOUT);
    }
    dim3 g2((N_NODES + 63) / 64, DOUT / 128);
    for (int r = 0; r < NREL; ++r) {
        gemm_kernel<DH, DOUT, true><<<g2, 256, 0, stream>>>(
            h1, W2 + (size_t)r * DH * DOUT, hW, N_NODES);
        scatter_kernel<DOUT><<<eblocks, 256, 0, stream>>>(
            hW, src + (size_t)r * NEDGE, dst + (size_t)r * NEDGE,
            c_src + (size_t)r * N_NODES, c_dst + (size_t)r * N_NODES, h2, NEDGE);
    }

    // ---- predictor: out[e] = dot(h2[src[et,e]], h2[dst[et,e]]) ----
    dot_kernel<<<eblocks, 256, 0, stream>>>(h2, src, dst, etype, out, NEDGE);
}